// CapsuleLayer_79216376807749
// MI455X (gfx1250) — compile-verified
//
#include <hip/hip_runtime.h>
#include <hip/hip_bf16.h>
#include <math.h>

// Problem constants (from reference setup_inputs)
#define BATCH   128
#define ICAP    1152   // input_num_capsule
#define DIN     8      // input_dim_capsule (contraction K)
#define NCAP    32     // num_capsule
#define DCAP    16     // dim_capsule
#define BT      16     // batch tile (WMMA M)
#define IC      48     // i's per chunk
#define NCHUNK  (ICAP / IC)   // 24
#define NWAVE   8
#define TPW     (NCAP / NWAVE) // 4 capsule-tiles per wave

typedef __attribute__((ext_vector_type(2))) float v2f;
typedef __attribute__((ext_vector_type(8))) float v8f;

// hat[b, n, i, d] = sum_k inputs[b,i,k] * W[n,i,d,k]
// Per i: GEMM M=batch-tile(16) x N=(n,d) x K=8 via 2x V_WMMA_F32_16X16X4_F32.
// PASS 0: c = 1/32 uniform, s0 = (1/32) * sum_i hat
// PASS 1: b1 = <out0, hat>_d  (write b_logits), c = softmax_n(b1), s1 = sum_i c*hat
// PASS 2: b2 = b1 + <out1, hat>_d (read b_logits), c = softmax_n(b2), s2 = sum_i c*hat
template <int PASS>
__global__ __launch_bounds__(256) void caps_pass_kernel(
    const float* __restrict__ inputs,   // [B, I, DIN]
    const float* __restrict__ W,        // [N, I, D, DIN]
    float* __restrict__ b_logits,       // [B, N, I]    (ws)
    float* __restrict__ s_part,         // [NCHUNK, B, N, D] (ws)
    const float* __restrict__ out_prev) // [B, N, D]    (ws)
{
    __shared__ float lds_b[NCAP][BT];   // logits for current i: [n][m]
    __shared__ float lds_max[BT];
    __shared__ float lds_rs[BT];

    const int chunk = blockIdx.x;       // 0..NCHUNK-1
    const int btile = blockIdx.y;       // 0..7
    const int b0    = btile * BT;
    const int i0    = chunk * IC;

    const int tid  = threadIdx.x;
    const int lane = tid & 31;
    const int wave = tid >> 5;
    const int row  = lane & 15;         // A row offset base / B column (d) / C column
    const int half = lane >> 4;         // 0 or 1 (upper K half for A/B, M+8 for C/D)
    const int klo  = half * 2;

    v8f s_acc[TPW];
#pragma unroll
    for (int tt = 0; tt < TPW; ++tt) s_acc[tt] = (v8f){};

    // Cache previous outputs: outp[tt][j] = out_prev[b0+m, n, d], m=j+half*8, d=row
    float outp[TPW][8];
    if constexpr (PASS > 0) {
#pragma unroll
        for (int tt = 0; tt < TPW; ++tt) {
            const int n = wave * TPW + tt;
#pragma unroll
            for (int j = 0; j < 8; ++j) {
                const int m = j + half * 8;
                outp[tt][j] = out_prev[(size_t)(b0 + m) * (NCAP * DCAP) + n * DCAP + row];
            }
        }
    }

    for (int ii = 0; ii < IC; ++ii) {
        const int i = i0 + ii;

        // A tile (16x8 f32, shared by all capsule tiles):
        // lanes 0-15: M=row, K={klo,klo+1}; lanes 16-31: K={2,3} (a0) / +4 (a1)
        const float* ap = inputs + (size_t)(b0 + row) * (ICAP * DIN) + (size_t)i * DIN;
        v2f a0; a0.x = ap[klo];     a0.y = ap[klo + 1];
        v2f a1; a1.x = ap[4 + klo]; a1.y = ap[4 + klo + 1];

        v8f acc[TPW];
#pragma unroll
        for (int tt = 0; tt < TPW; ++tt) {
            const int n = wave * TPW + tt;
            // B tile (8 x 16 cols = d): lane is column d, K striped across the 2 VGPRs
            const float* wp = W + (size_t)n * (ICAP * DCAP * DIN)
                                + (size_t)i * (DCAP * DIN) + (size_t)row * DIN;
            v2f bm0; bm0.x = wp[klo];     bm0.y = wp[klo + 1];
            v2f bm1; bm1.x = wp[4 + klo]; bm1.y = wp[4 + klo + 1];

            v8f c = (v8f){};
            c = __builtin_amdgcn_wmma_f32_16x16x4_f32(false, a0, false, bm0,
                                                      (short)0, c, false, false);
            c = __builtin_amdgcn_wmma_f32_16x16x4_f32(false, a1, false, bm1,
                                                      (short)0, c, false, false);
            acc[tt] = c;
        }

        if constexpr (PASS == 0) {
#pragma unroll
            for (int tt = 0; tt < TPW; ++tt) s_acc[tt] += acc[tt];
        } else {
            // Agreement: g[m] = sum_d out_prev[b,n,d] * hat[b,n,i,d]
#pragma unroll
            for (int tt = 0; tt < TPW; ++tt) {
                const int n = wave * TPW + tt;
                float g[8];
#pragma unroll
                for (int j = 0; j < 8; ++j) {
                    float p = acc[tt][j] * outp[tt][j];
                    p += __shfl_xor(p, 1, 16);
                    p += __shfl_xor(p, 2, 16);
                    p += __shfl_xor(p, 4, 16);
                    p += __shfl_xor(p, 8, 16);
                    g[j] = p;
                }
                if (row == 0) {  // lane 0 -> m=0..7, lane 16 -> m=8..15
#pragma unroll
                    for (int j = 0; j < 8; ++j) {
                        const int m = j + half * 8;
                        const size_t bidx = (size_t)(b0 + m) * (NCAP * (size_t)ICAP)
                                          + (size_t)n * ICAP + i;
                        float bl = g[j];
                        if constexpr (PASS == 2) bl += b_logits[bidx];
                        else                     b_logits[bidx] = bl;
                        lds_b[n][m] = bl;
                    }
                }
            }
            __syncthreads();
            // softmax over n (32 values) per m
            if (tid < BT) {
                const int m = tid;
                float mx = -3.402823466e38f;
                for (int n = 0; n < NCAP; ++n) mx = fmaxf(mx, lds_b[n][m]);
                float sum = 0.0f;
                for (int n = 0; n < NCAP; ++n) sum += __expf(lds_b[n][m] - mx);
                lds_max[m] = mx;
                lds_rs[m]  = 1.0f / sum;
            }
            __syncthreads();
#pragma unroll
            for (int tt = 0; tt < TPW; ++tt) {
                const int n = wave * TPW + tt;
#pragma unroll
                for (int j = 0; j < 8; ++j) {
                    const int m = j + half * 8;
                    const float c = __expf(lds_b[n][m] - lds_max[m]) * lds_rs[m];
                    s_acc[tt][j] += c * acc[tt][j];
                }
            }
            __syncthreads();  // lds_b reused next i
        }
    }

    // Store partial s for this chunk: s_part[chunk, b, n, d]
    const float scale = (PASS == 0) ? (1.0f / NCAP) : 1.0f;
#pragma unroll
    for (int tt = 0; tt < TPW; ++tt) {
        const int n = wave * TPW + tt;
#pragma unroll
        for (int j = 0; j < 8; ++j) {
            const int m = j + half * 8;
            s_part[((size_t)chunk * BATCH + b0 + m) * (NCAP * DCAP) + n * DCAP + row]
                = s_acc[tt][j] * scale;
        }
    }
}

// Reduce chunk partials (fixed order -> deterministic) and apply squash.
__global__ __launch_bounds__(256) void caps_squash_kernel(
    const float* __restrict__ s_part, float* __restrict__ out)
{
    const int idx = blockIdx.x * 256 + threadIdx.x;  // over (b*512 + n*16 + d)
    float s = 0.0f;
    for (int ch = 0; ch < NCHUNK; ++ch)
        s += s_part[(size_t)ch * (BATCH * NCAP * DCAP) + idx];
    float sq = s * s;                   // d = idx & 15 aligns with 16-lane groups
    sq += __shfl_xor(sq, 1, 16);
    sq += __shfl_xor(sq, 2, 16);
    sq += __shfl_xor(sq, 4, 16);
    sq += __shfl_xor(sq, 8, 16);
    const float scale = sq / ((1.0f + sq) * sqrtf(sq + 1e-7f));
    out[idx] = scale * s;
}

extern "C" void kernel_launch(void* const* d_in, const int* in_sizes, int n_in,
                              void* d_out, int out_size, void* d_ws, size_t ws_size,
                              hipStream_t stream) {
    const float* inputs = (const float*)d_in[0];  // [128,1152,8]
    const float* W      = (const float*)d_in[1];  // [32,1152,16,8]
    float* out = (float*)d_out;                   // [128,32,16]

    char* ws = (char*)d_ws;
    float* b_logits = (float*)ws;                                     // 18,874,368 B
    float* s_part   = (float*)(ws + (size_t)BATCH * NCAP * ICAP * 4); //  6,291,456 B
    float* out_prev = (float*)(ws + (size_t)BATCH * NCAP * ICAP * 4
                                  + (size_t)NCHUNK * BATCH * NCAP * DCAP * 4); // 262,144 B

    dim3 grid(NCHUNK, BATCH / BT);
    dim3 blk(256);
    const int sqg = (BATCH * NCAP * DCAP) / 256;

    caps_pass_kernel<0><<<grid, blk, 0, stream>>>(inputs, W, b_logits, s_part, out_prev);
    caps_squash_kernel<<<sqg, 256, 0, stream>>>(s_part, out_prev);
    caps_pass_kernel<1><<<grid, blk, 0, stream>>>(inputs, W, b_logits, s_part, out_prev);
    caps_squash_kernel<<<sqg, 256, 0, stream>>>(s_part, out_prev);
    caps_pass_kernel<2><<<grid, blk, 0, stream>>>(inputs, W, b_logits, s_part, out_prev);
    caps_squash_kernel<<<sqg, 256, 0, stream>>>(s_part, out);
}